// DeepMmdLoss_3770981286253
// MI455X (gfx1250) — compile-verified
//
#include <hip/hip_runtime.h>
#include <hip/hip_bf16.h>
#include <math.h>

// Problem constants (from reference)
#define LEN_SZ 4096
#define IN_DIM 256
#define HID    10
#define OUT_F  50
#define FPAD   64   // feats padded to 64 so K-loop is a multiple of 4

// LDS row pitches (floats). pitch % 64 == 6 -> conflict-free 16-lane row reads,
// and pitch*4 % 8 == 0 -> aligned ds_load_b64.
#define PITCH_O 262   // 256 + 6 pad dwords (TDM pad: interval=256 dw, amount=6 dw)
#define PITCH_F 70    // 64 + 6 pad dwords  (TDM pad: interval=64 dw,  amount=6 dw)

typedef __attribute__((ext_vector_type(2))) float v2f;
typedef __attribute__((ext_vector_type(8))) float v8f;
typedef __attribute__((ext_vector_type(4))) unsigned int v4u;
typedef __attribute__((ext_vector_type(4))) int v4i;
typedef __attribute__((ext_vector_type(8))) int v8i;

// ---------------------------------------------------------------------------
// fp32 WMMA: D(16x16) += A(16x4) * B(4x16), full precision (CDNA5)
// ---------------------------------------------------------------------------
__device__ __forceinline__ v8f wmma_f32(v2f a, v2f b, v8f c) {
  // 8 args: (neg_a, A, neg_b, B, c_mod, C, reuse_a, reuse_b)
  return __builtin_amdgcn_wmma_f32_16x16x4_f32(false, a, false, b, (short)0, c,
                                               false, false);
}

// B fragment load (4x16 fp32) from a row-major global matrix.
// lane<16: row = row0+lane, K = k..k+1 ; lane>=16: row = row0+lane-16, K = k+2..k+3
__device__ __forceinline__ v2f load_frag(const float* __restrict__ base, int row0,
                                         int stride, int k, int l16, int half) {
  const float2* p =
      (const float2*)(base + (size_t)(row0 + l16) * stride + k + 2 * half);
  float2 t = *p;
  v2f r;
  r[0] = t.x;
  r[1] = t.y;
  return r;
}

// A fragment load (16x4 fp32) from an LDS-staged tile with padded pitch.
__device__ __forceinline__ v2f lds_frag(const float* base, int pitch, int k,
                                        int l16, int half) {
  const float2* p = (const float2*)(base + l16 * pitch + k + 2 * half);
  float2 t = *p;
  v2f r;
  r[0] = t.x;
  r[1] = t.y;
  return r;
}

// ---------------------------------------------------------------------------
// Tensor Data Mover: 2-D tile load Global->LDS with row padding.
// D# packed per CDNA5 ISA (group0 128b, group1 256b).
// width/rows in 4-byte elements; pad_code: pad_interval enum; pad_dw_m1: dwords-1.
// This toolchain exposes the 6-arg builtin form.
// ---------------------------------------------------------------------------
__device__ __forceinline__ void tdm_load_2d(const float* gsrc, unsigned lds_off,
                                            unsigned width, unsigned rows,
                                            unsigned pad_code, unsigned pad_dw_m1) {
  unsigned long long ga = (unsigned long long)(const void*)gsrc;
  v4u g0;
  g0[0] = 1u;  // count=1 (valid user descriptor), gather off, not restore
  g0[1] = lds_off;
  g0[2] = (unsigned)(ga & 0xffffffffu);
  g0[3] = (unsigned)((ga >> 32) & 0x01ffffffu) | (2u << 30);  // type=2 ("image")
  v8i g1;
  unsigned flags = (2u << 16)           // data_size = 4 bytes
                   | (1u << 20)         // pad_enable
                   | (pad_code << 22)   // pad_interval
                   | (pad_dw_m1 << 25); // pad_amount
  g1[0] = (int)flags;                               // mask=0, no atomic barrier
  g1[1] = (int)((width & 0xffffu) << 16);           // tensor_dim0[15:0]
  g1[2] = (int)((width >> 16) | ((rows & 0xffffu) << 16));  // dim0 hi | dim1 lo
  g1[3] = (int)((rows >> 16) | (width << 16));      // dim1 hi | tile_dim0
  g1[4] = (int)(rows & 0xffffu);                    // tile_dim1 | tile_dim2=0
  g1[5] = (int)width;                               // tensor_dim0_stride lo
  g1[6] = 0;                                        // stride0 hi | stride1 lo
  g1[7] = 0;                                        // stride1 hi
  v4i z4 = {0, 0, 0, 0};                            // 2-D: groups 2/3 unused
  v8i z8 = {0, 0, 0, 0, 0, 0, 0, 0};
  __builtin_amdgcn_tensor_load_to_lds(g0, g1, z4, z4, z8, 0);
}

__device__ __forceinline__ float softplus_f(float x) {
  return fmaxf(x, 0.f) + log1pf(expf(-fabsf(x)));
}

// ---------------------------------------------------------------------------
// Zero-init accumulator scratch (harness poisons d_ws with 0xAA)
// ---------------------------------------------------------------------------
__global__ void zero_kernel(float* __restrict__ p, int n) {
  int i = blockIdx.x * blockDim.x + threadIdx.x;
  if (i < n) p[i] = 0.f;
}

// ---------------------------------------------------------------------------
// MLP (256->10->10->10->50) + squared norms of both spaces.
// ---------------------------------------------------------------------------
__global__ void mlp_kernel(const float* __restrict__ X, const float* __restrict__ Y,
                           const float* __restrict__ W1, const float* __restrict__ b1,
                           const float* __restrict__ W2, const float* __restrict__ b2,
                           const float* __restrict__ W3, const float* __restrict__ b3,
                           const float* __restrict__ W4, const float* __restrict__ b4,
                           float* __restrict__ feats, float* __restrict__ normf,
                           float* __restrict__ normo) {
  int row = blockIdx.x * blockDim.x + threadIdx.x;
  if (row >= 2 * LEN_SZ) return;
  const float* src = (row < LEN_SZ) ? (X + (size_t)row * IN_DIM)
                                    : (Y + (size_t)(row - LEN_SZ) * IN_DIM);
  float h1[HID];
#pragma unroll
  for (int j = 0; j < HID; ++j) h1[j] = b1[j];
  float no = 0.f;
  for (int k = 0; k < IN_DIM; ++k) {
    float xv = src[k];
    no += xv * xv;
    const float* w = W1 + k * HID;
#pragma unroll
    for (int j = 0; j < HID; ++j) h1[j] = fmaf(xv, w[j], h1[j]);
  }
#pragma unroll
  for (int j = 0; j < HID; ++j) h1[j] = softplus_f(h1[j]);

  float h2[HID];
#pragma unroll
  for (int j = 0; j < HID; ++j) h2[j] = b2[j];
#pragma unroll
  for (int k = 0; k < HID; ++k) {
    float hv = h1[k];
    const float* w = W2 + k * HID;
#pragma unroll
    for (int j = 0; j < HID; ++j) h2[j] = fmaf(hv, w[j], h2[j]);
  }
#pragma unroll
  for (int j = 0; j < HID; ++j) h2[j] = softplus_f(h2[j]);

  float h3[HID];
#pragma unroll
  for (int j = 0; j < HID; ++j) h3[j] = b3[j];
#pragma unroll
  for (int k = 0; k < HID; ++k) {
    float hv = h2[k];
    const float* w = W3 + k * HID;
#pragma unroll
    for (int j = 0; j < HID; ++j) h3[j] = fmaf(hv, w[j], h3[j]);
  }
#pragma unroll
  for (int j = 0; j < HID; ++j) h3[j] = softplus_f(h3[j]);

  float nf = 0.f;
  float* dst = feats + (size_t)row * FPAD;
  for (int j = 0; j < OUT_F; ++j) {
    float o = b4[j];
#pragma unroll
    for (int k = 0; k < HID; ++k) o = fmaf(h3[k], W4[k * OUT_F + j], o);
    nf += o * o;
    dst[j] = o;
  }
#pragma unroll
  for (int j = OUT_F; j < FPAD; ++j) dst[j] = 0.f;  // zero padding
  normf[row] = nf;
  normo[row] = no;
}

// ---------------------------------------------------------------------------
// Main kernel: block = tile-row tI x 4 consecutive tJ (one 16x16 tile/wave).
// A-side tiles (rows i0..i0+15 of X, Y, feats_x, feats_y) are staged once per
// block into LDS by the Tensor Data Mover (padded pitch -> conflict-free reads),
// B-side fragments stream from global/L2. Six fp32 WMMA accumulators per wave.
// ---------------------------------------------------------------------------
__global__ void __launch_bounds__(128)
mmd_tile_kernel(const float* __restrict__ X, const float* __restrict__ Y,
                const float* __restrict__ feats, const float* __restrict__ normf,
                const float* __restrict__ normo, const float* __restrict__ peps,
                const float* __restrict__ psq, const float* __restrict__ psp,
                float* __restrict__ hrow, float* __restrict__ kcol,
                float* __restrict__ S) {
  __shared__ float ldsX[16 * PITCH_O];
  __shared__ float ldsY[16 * PITCH_O];
  __shared__ float ldsFX[16 * PITCH_F];
  __shared__ float ldsFY[16 * PITCH_F];

  const int lane = threadIdx.x & 31;
  const int half = lane >> 4;
  const int l16 = lane & 15;
  const int waveInBlk = threadIdx.x >> 5;
  const int tI = blockIdx.x >> 6;                       // 256 tile-rows
  const int tJ = ((blockIdx.x & 63) << 2) | waveInBlk;  // 4 J-tiles per block
  const int i0 = tI * 16, j0 = tJ * 16;

  const float* fx = feats;
  const float* fy = feats + (size_t)LEN_SZ * FPAD;

  // Wave 0 issues the four TDM tile loads for the shared A-side rows.
  if (waveInBlk == 0) {
    unsigned offX  = (unsigned)(size_t)&ldsX[0];
    unsigned offY  = (unsigned)(size_t)&ldsY[0];
    unsigned offFX = (unsigned)(size_t)&ldsFX[0];
    unsigned offFY = (unsigned)(size_t)&ldsFY[0];
    // orig rows: 16 x 256 f32, pad 6 dw every 256 dw (code 7) -> pitch 262
    tdm_load_2d(X + (size_t)i0 * IN_DIM, offX, IN_DIM, 16, 7u, 5u);
    tdm_load_2d(Y + (size_t)i0 * IN_DIM, offY, IN_DIM, 16, 7u, 5u);
    // feat rows: 16 x 64 f32, pad 6 dw every 64 dw (code 5) -> pitch 70
    tdm_load_2d(fx + (size_t)i0 * FPAD, offFX, FPAD, 16, 5u, 5u);
    tdm_load_2d(fy + (size_t)i0 * FPAD, offFY, FPAD, 16, 5u, 5u);
    __builtin_amdgcn_s_wait_tensorcnt(0);
  }
  __syncthreads();

  v8f aox = {};  // xo . xo
  v8f aoy = {};  // yo . yo
  v8f aoz = {};  // xo . yo
  v8f afx = {};  // x . x     (feature space)
  v8f afy = {};  // y . y
  v8f afz = {};  // x . y

  // Original 256-dim space: 64 K-steps of 4 (A from LDS, B from global/L2)
  for (int k = 0; k < IN_DIM; k += 4) {
    v2f ax = lds_frag(ldsX, PITCH_O, k, l16, half);
    v2f ay = lds_frag(ldsY, PITCH_O, k, l16, half);
    v2f bx = load_frag(X, j0, IN_DIM, k, l16, half);
    v2f by = load_frag(Y, j0, IN_DIM, k, l16, half);
    aox = wmma_f32(ax, bx, aox);
    aoy = wmma_f32(ay, by, aoy);
    aoz = wmma_f32(ax, by, aoz);
  }
  // Feature space (50-dim padded to 64): 16 K-steps of 4
  for (int k = 0; k < FPAD; k += 4) {
    v2f ax = lds_frag(ldsFX, PITCH_F, k, l16, half);
    v2f ay = lds_frag(ldsFY, PITCH_F, k, l16, half);
    v2f bx = load_frag(fx, j0, FPAD, k, l16, half);
    v2f by = load_frag(fy, j0, FPAD, k, l16, half);
    afx = wmma_f32(ax, bx, afx);
    afy = wmma_f32(ay, by, afy);
    afz = wmma_f32(ax, by, afz);
  }

  const float eps = 1.f / (1.f + expf(-peps[0]));
  const float inv_sq = 1.f / (psq[0] * psq[0]);
  const float inv_sp = 1.f / (psp[0] * psp[0]);
  const float ome = 1.f - eps;

  // j-side norms: column index = j0 + l16 (same for both lane halves)
  const float nfx_j = normf[j0 + l16];
  const float nfy_j = normf[LEN_SZ + j0 + l16];
  const float nox_j = normo[j0 + l16];
  const float noy_j = normo[LEN_SZ + j0 + l16];

  const bool diag = (tI == tJ);
  float skx = 0.f, sky = 0.f, skz = 0.f, col = 0.f;
  float tkx = 0.f, tky = 0.f, tkz = 0.f;

#pragma unroll
  for (int r = 0; r < 8; ++r) {
    const int mi = i0 + r + 8 * half;  // global row index for this VGPR slot
    const float nfx_i = normf[mi];
    const float nfy_i = normf[LEN_SZ + mi];
    const float nox_i = normo[mi];
    const float noy_i = normo[LEN_SZ + mi];

    float dfx = fmaxf(nfx_i + nfx_j - 2.f * afx[r], 0.f);
    float dox = fmaxf(nox_i + nox_j - 2.f * aox[r], 0.f);
    float dfy = fmaxf(nfy_i + nfy_j - 2.f * afy[r], 0.f);
    float doy = fmaxf(noy_i + noy_j - 2.f * aoy[r], 0.f);
    float dfz = fmaxf(nfx_i + nfy_j - 2.f * afz[r], 0.f);
    float doz = fmaxf(nox_i + noy_j - 2.f * aoz[r], 0.f);

    // k = exp(-do/sq) * ((1-eps)*exp(-d/sp) + eps)
    float ex = expf(-dox * inv_sq);
    float ey = expf(-doy * inv_sq);
    float ez = expf(-doz * inv_sq);
    float kxv = ex * (ome * expf(-dfx * inv_sp) + eps);
    float kyv = ey * (ome * expf(-dfy * inv_sp) + eps);
    float kzv = ez * (ome * expf(-dfz * inv_sp) + eps);

    skx += kxv;
    sky += kyv;
    skz += kzv;
    col += kzv;  // k_xy column partial

    // row sum of (k_x + k_y - k_xy) across the 16 N-lanes of each half
    float q = kxv + kyv - kzv;
#pragma unroll
    for (int m = 1; m <= 8; m <<= 1) q += __shfl_xor(q, m, 32);
    if (l16 == 0) atomicAdd(&hrow[mi], q);  // lanes 0 and 16 -> rows r, r+8

    if (diag) {
      // diagonal element (m==n): lane==r (half 0) or lane==r+24 (half 1)
      bool sel = (lane == r) | (lane == r + 24);
      tkx += sel ? kxv : 0.f;
      tky += sel ? kyv : 0.f;
      tkz += sel ? kzv : 0.f;
    }
  }

  // k_xy column sums: combine the two lane halves, lanes 0..15 commit
  col += __shfl_xor(col, 16, 32);
  if (half == 0) atomicAdd(&kcol[j0 + l16], col);

#pragma unroll
  for (int m = 1; m <= 16; m <<= 1) {
    skx += __shfl_xor(skx, m, 32);
    sky += __shfl_xor(sky, m, 32);
    skz += __shfl_xor(skz, m, 32);
  }
  if (lane == 0) {
    atomicAdd(&S[0], skx);
    atomicAdd(&S[1], sky);
    atomicAdd(&S[2], skz);
  }
  if (diag) {
#pragma unroll
    for (int m = 1; m <= 16; m <<= 1) {
      tkx += __shfl_xor(tkx, m, 32);
      tky += __shfl_xor(tky, m, 32);
      tkz += __shfl_xor(tkz, m, 32);
    }
    if (lane == 0) {
      atomicAdd(&S[3], tkx);
      atomicAdd(&S[4], tky);
      atomicAdd(&S[5], tkz);
    }
  }
}

// ---------------------------------------------------------------------------
// Finalize: hs[i] = hrow[i] - kxy_col[i]; produce [mmd2, var]
// ---------------------------------------------------------------------------
__global__ void finalize_kernel(const float* __restrict__ hrow,
                                const float* __restrict__ kcol,
                                const float* __restrict__ S,
                                float* __restrict__ out) {
  __shared__ float sh_s[256];
  __shared__ float sh_d[256];
  const int t = threadIdx.x;
  float s = 0.f, d = 0.f;
  for (int i = t; i < LEN_SZ; i += 256) {
    float v = hrow[i] - kcol[i];
    s += v;
    d += v * v;
  }
  sh_s[t] = s;
  sh_d[t] = d;
  __syncthreads();
  for (int off = 128; off > 0; off >>= 1) {
    if (t < off) {
      sh_s[t] += sh_s[t + off];
      sh_d[t] += sh_d[t + off];
    }
    __syncthreads();
  }
  if (t == 0) {
    const double n = (double)LEN_SZ;
    const float denom = (float)(n * (n - 1.0));
    float xx = (S[0] - S[3]) / denom;
    float yy = (S[1] - S[4]) / denom;
    float xy = (S[2] - S[5]) / denom;
    out[0] = xx - 2.f * xy + yy;
    float hs_sum = sh_s[0], hs_dot = sh_d[0];
    float v1 = (float)(4.0 / (n * n * n)) * hs_dot;
    float v2 = (float)(4.0 / (n * n * n * n)) * hs_sum * hs_sum;
    out[1] = v1 - v2 + 1e-8f;
  }
}

// ---------------------------------------------------------------------------
extern "C" void kernel_launch(void* const* d_in, const int* in_sizes, int n_in,
                              void* d_out, int out_size, void* d_ws, size_t ws_size,
                              hipStream_t stream) {
  const float* X = (const float*)d_in[0];
  const float* Y = (const float*)d_in[1];
  const float* W1 = (const float*)d_in[2];
  const float* b1 = (const float*)d_in[3];
  const float* W2 = (const float*)d_in[4];
  const float* b2 = (const float*)d_in[5];
  const float* W3 = (const float*)d_in[6];
  const float* b3 = (const float*)d_in[7];
  const float* W4 = (const float*)d_in[8];
  const float* b4 = (const float*)d_in[9];
  const float* peps = (const float*)d_in[10];
  const float* psq = (const float*)d_in[11];
  const float* psp = (const float*)d_in[12];
  float* out = (float*)d_out;

  // Workspace layout (floats): feats[8192*64] | normf[8192] | normo[8192]
  //                            | hrow[4096] | kcol[4096] | S[8]
  float* feats = (float*)d_ws;
  float* normf = feats + (size_t)2 * LEN_SZ * FPAD;
  float* normo = normf + 2 * LEN_SZ;
  float* hrow = normo + 2 * LEN_SZ;
  float* kcol = hrow + LEN_SZ;
  float* S = kcol + LEN_SZ;

  // hrow, kcol, S are contiguous: zero them every call (graph-replay safe)
  const int nzero = 2 * LEN_SZ + 8;
  zero_kernel<<<(nzero + 255) / 256, 256, 0, stream>>>(hrow, nzero);

  mlp_kernel<<<(2 * LEN_SZ + 255) / 256, 256, 0, stream>>>(
      X, Y, W1, b1, W2, b2, W3, b3, W4, b4, feats, normf, normo);

  // 256 tile-rows x 64 blocks, each block = 4 waves = 4 J-tiles sharing
  // TDM-staged A-side LDS tiles.
  mmd_tile_kernel<<<256 * 64, 128, 0, stream>>>(
      X, Y, feats, normf, normo, peps, psq, psp, hrow, kcol, S);

  finalize_kernel<<<1, 256, 0, stream>>>(hrow, kcol, S, out);
}